// AdaptiveDeformedFusion_36601711296533
// MI455X (gfx1250) — compile-verified
//
#include <hip/hip_runtime.h>

// ---------------------------------------------------------------------------
// AdaptiveDeformedFusion for MI455X (gfx1250, wave32).
// GEMM-shaped ops -> WMMA f32 16x16x4 (full precision; workload is memory-
// bound: ~600MB unique traffic @ 23.3TB/s ~= 26us floor, so no quantization).
// Tile staging -> Tensor Data Mover (tensor_load_to_lds, TENSORcnt) where the
// tile is a dense 2D slice; im2col gathers stay on per-lane VMEM.
// ---------------------------------------------------------------------------

typedef __attribute__((ext_vector_type(2))) float v2f;
typedef __attribute__((ext_vector_type(8))) float v8f;
typedef __attribute__((ext_vector_type(4))) unsigned int v4u;
typedef __attribute__((ext_vector_type(8))) int v8i;
typedef __attribute__((ext_vector_type(4))) int v4i;

#if __has_builtin(__builtin_amdgcn_tensor_load_to_lds) && \
    __has_builtin(__builtin_amdgcn_s_wait_tensorcnt)
#define HAVE_TDM 1
#else
#define HAVE_TDM 0
#endif

__device__ __forceinline__ v8f wmma_f32_4(v2f a, v2f b, v8f c) {
  return __builtin_amdgcn_wmma_f32_16x16x4_f32(
      /*neg_a=*/false, a, /*neg_b=*/false, b,
      /*c_mod=*/(short)0, c, /*reuse_a=*/false, /*reuse_b=*/false);
}

namespace {

constexpr int BATCH = 4;
constexpr int HH = 80, WW = 80, HW = HH * WW;   // 6400
constexpr int INC = 256;                         // IN_CH
constexpr int DIMS = 128;

#if HAVE_TDM
// 2D TDM load: tile (tile_y rows x tile_x elems, f32) from a row-major tensor
// with row stride stride_x (elems); remaining extents rem_x/rem_y give HW
// zero-fill for out-of-tensor reads. pad_iv encodes LDS padding interval
// (3 => +1 DWORD per 16 DWORDs, 4 => per 32 DWORDs) to keep bank-conflict
// padded LDS layouts ([r][16+1] / [r][32+1]).
__device__ __forceinline__ void tdm_load_2d(unsigned lds_addr, const float* gp,
                                            unsigned rem_x, unsigned rem_y,
                                            unsigned tile_x, unsigned tile_y,
                                            unsigned long long stride_x,
                                            unsigned pad_iv) {
  unsigned long long ga = (unsigned long long)(size_t)gp;
  v4u g0;
  g0.x = 1u;                                          // count=1, user mode
  g0.y = lds_addr;                                    // LDS byte address
  g0.z = (unsigned)(ga & 0xffffffffu);                // global_addr[31:0]
  g0.w = (unsigned)((ga >> 32) & 0x01ffffffu) | (2u << 30);  // [56:32] | type=2
  unsigned long long sx = stride_x;                   // elem units (data_size)
  v8i g1;
  g1[0] = (int)((2u << 16) | (1u << 20) | (pad_iv << 22));  // f32, pad_en, iv, amt=1dw
  g1[1] = (int)((rem_x & 0xffffu) << 16);                   // tensor_dim0 lo
  g1[2] = (int)((rem_x >> 16) | ((rem_y & 0xffffu) << 16)); // dim0 hi | dim1 lo
  g1[3] = (int)((rem_y >> 16) | (tile_x << 16));            // dim1 hi | tile_dim0
  g1[4] = (int)(tile_y & 0xffffu);                          // tile_dim1, tile_dim2=0
  g1[5] = (int)(unsigned)(sx & 0xffffffffu);                // dim0_stride lo
  g1[6] = (int)(unsigned)((sx >> 32) & 0xffffu);            // dim0_stride hi
  g1[7] = 0;                                                // dim1_stride (2D)
  v4i z4 = {0, 0, 0, 0};
#if defined(__clang_major__) && __clang_major__ >= 23
  v8i z8 = {0, 0, 0, 0, 0, 0, 0, 0};
  __builtin_amdgcn_tensor_load_to_lds(g0, g1, z4, z4, z8, 0);
#else
  __builtin_amdgcn_tensor_load_to_lds(g0, g1, z4, z4, 0);
#endif
}
#endif  // HAVE_TDM

// ------------------------- gather functors (implicit im2col) ---------------

// 3x3 conv, stride 1, pad 1 over a 2-source channel concat.
struct GConv3Pad {
  const float* S0; size_t s0bs; int c0;
  const float* S1; size_t s1bs;
  int H, W;
  __device__ float operator()(int b, int k, int m) const {
    int c = k / 9, t = k - c * 9;
    int y = m / W + t / 3 - 1;
    int x = m % W + t % 3 - 1;
    if ((unsigned)y >= (unsigned)H || (unsigned)x >= (unsigned)W) return 0.f;
    const float* s = (c < c0) ? S0 + (size_t)b * s0bs + (size_t)c * H * W
                              : S1 + (size_t)b * s1bs + (size_t)(c - c0) * H * W;
    return s[y * W + x];
  }
};

// KSxKS conv, stride 2, pad, over up to 3-source channel concat.
struct GConvS2 {
  const float* S0; size_t s0bs; int c0;
  const float* S1; size_t s1bs; int c1n;
  const float* S2; size_t s2bs;
  int KS, pad, Hin, Win, Wout;
  __device__ float operator()(int b, int k, int m) const {
    const int ksq = KS * KS;
    int c = k / ksq, t = k - c * ksq;
    int ym = m / Wout, xm = m - ym * Wout;
    int y = 2 * ym + t / KS - pad;
    int x = 2 * xm + t % KS - pad;
    if ((unsigned)y >= (unsigned)Hin || (unsigned)x >= (unsigned)Win) return 0.f;
    const float* s; int cc;
    if (c < c0)            { s = S0 + (size_t)b * s0bs; cc = c; }
    else if (c < c0 + c1n) { s = S1 + (size_t)b * s1bs; cc = c - c0; }
    else                   { s = S2 + (size_t)b * s2bs; cc = c - c0 - c1n; }
    return s[(size_t)cc * Hin * Win + y * Win + x];
  }
};

// ------------------------- shared WMMA helpers ------------------------------

// One K=16 chunk: 4x v_wmma_f32_16x16x4_f32 from padded LDS tiles.
__device__ __forceinline__ void wmma_chunk(const float (*sW)[17], const float (*sX)[33],
                                           int nsub, int msub, int lane, v8f& acc) {
  const int row = lane & 15;
  const int kh = (lane >> 4) << 1;  // 0 or 2 (f32 A/B lane-half K split)
#pragma unroll
  for (int ks = 0; ks < 16; ks += 4) {
    v2f af, bf;
    af.x = sW[nsub + row][ks + kh];
    af.y = sW[nsub + row][ks + kh + 1];
    bf.x = sX[ks + kh][msub + row];
    bf.y = sX[ks + kh + 1][msub + row];
    acc = wmma_f32_4(af, bf, acc);
  }
}

template <bool RELU>
__device__ __forceinline__ void wmma_epilogue(v8f acc, float* Ob, const float* Bb,
                                              int n0, int m0, int nsub, int msub,
                                              int lane, int N, int M) {
  const int mcol = lane & 15;
  const int nhalf = (lane >> 4) * 8;
#pragma unroll
  for (int r = 0; r < 8; ++r) {
    int n = n0 + nsub + r + nhalf;
    int m = m0 + msub + mcol;
    if (n < N && m < M) {
      float v = acc[r];
      if (Bb) v += Bb[n];
      if (RELU) v = fmaxf(v, 0.f);
      Ob[(size_t)n * M + m] = v;
    }
  }
}

// ------------------ WMMA GEMM, plain X[K][M]: full TDM staging --------------
// Out[b][n][m] = sum_k W[b][n][k] * X[b][k][m] + bias[b][n]
template <bool RELU>
__global__ __launch_bounds__(256) void gemm_wmma_plain(
    const float* __restrict__ W, size_t wbs,
    const float* __restrict__ X, size_t xbs,
    const float* __restrict__ bias, size_t bias_bs,
    float* __restrict__ Out, size_t out_bs,
    int N, int K, int M) {
  __shared__ float sW[64][17];
  __shared__ float sX[16][33];
  const int b = blockIdx.z;
  const int n0 = blockIdx.y * 64;
  const int m0 = blockIdx.x * 32;
  const int tid = threadIdx.x;
  const int wave = tid >> 5;
  const int lane = tid & 31;
  const int nsub = (wave & 3) * 16;
  const int msub = (wave >> 2) * 16;
  const float* Wb = W + (size_t)b * wbs;
  const float* Xb = X + (size_t)b * xbs;

  v8f acc = {0.f, 0.f, 0.f, 0.f, 0.f, 0.f, 0.f, 0.f};
#if HAVE_TDM
  const unsigned ldsW = (unsigned)(size_t)&sW[0][0];
  const unsigned ldsX = (unsigned)(size_t)&sX[0][0];
#endif

  for (int k0 = 0; k0 < K; k0 += 16) {
#if HAVE_TDM
    if (wave == 0) {
      // W tile: 64 rows x 16 k, row stride K; zero-fill beyond (N-n0, K-k0).
      tdm_load_2d(ldsW, Wb + (size_t)n0 * K + k0,
                  (unsigned)(K - k0), (unsigned)(N - n0), 16u, 64u,
                  (unsigned long long)K, 3u);
      // X tile: 16 rows x 32 m, row stride M; zero-fill beyond (K-k0, M-m0).
      tdm_load_2d(ldsX, Xb + (size_t)k0 * M + m0,
                  (unsigned)(M - m0), (unsigned)(K - k0), 32u, 16u,
                  (unsigned long long)M, 4u);
      __builtin_amdgcn_s_wait_tensorcnt(0);
    }
#else
    for (int i = tid; i < 64 * 16; i += 256) {
      int n = i >> 4, kk = i & 15;
      int gn = n0 + n, gk = k0 + kk;
      sW[n][kk] = (gn < N && gk < K) ? Wb[(size_t)gn * K + gk] : 0.f;
    }
    for (int i = tid; i < 16 * 32; i += 256) {
      int kk = i >> 5, mm = i & 31;
      int gk = k0 + kk, gm = m0 + mm;
      sX[kk][mm] = (gk < K && gm < M) ? Xb[(size_t)gk * M + gm] : 0.f;
    }
#endif
    __syncthreads();
    wmma_chunk(sW, sX, nsub, msub, lane, acc);
    __syncthreads();
  }
  wmma_epilogue<RELU>(acc, Out + (size_t)b * out_bs,
                      bias ? bias + (size_t)b * bias_bs : nullptr,
                      n0, m0, nsub, msub, lane, N, M);
}

// --------- WMMA GEMM, functor X (im2col): TDM for W, lane gather for X ------
template <class G, bool RELU>
__global__ __launch_bounds__(256) void gemm_wmma_g(
    const float* __restrict__ W, size_t wbs,
    const float* __restrict__ bias, size_t bias_bs,
    float* __restrict__ Out, size_t out_bs,
    int N, int K, int M, G g) {
  __shared__ float sW[64][17];
  __shared__ float sX[16][33];
  const int b = blockIdx.z;
  const int n0 = blockIdx.y * 64;
  const int m0 = blockIdx.x * 32;
  const int tid = threadIdx.x;
  const int wave = tid >> 5;
  const int lane = tid & 31;
  const int nsub = (wave & 3) * 16;
  const int msub = (wave >> 2) * 16;
  const float* Wb = W + (size_t)b * wbs;

  v8f acc = {0.f, 0.f, 0.f, 0.f, 0.f, 0.f, 0.f, 0.f};
#if HAVE_TDM
  const unsigned ldsW = (unsigned)(size_t)&sW[0][0];
#endif

  for (int k0 = 0; k0 < K; k0 += 16) {
#if HAVE_TDM
    if (wave == 0) {
      tdm_load_2d(ldsW, Wb + (size_t)n0 * K + k0,
                  (unsigned)(K - k0), (unsigned)(N - n0), 16u, 64u,
                  (unsigned long long)K, 3u);
    }
#else
    for (int i = tid; i < 64 * 16; i += 256) {
      int n = i >> 4, kk = i & 15;
      int gn = n0 + n, gk = k0 + kk;
      sW[n][kk] = (gn < N && gk < K) ? Wb[(size_t)gn * K + gk] : 0.f;
    }
#endif
    // im2col gather for X tile (padding/multi-source: not TDM-expressible)
    for (int i = tid; i < 16 * 32; i += 256) {
      int kk = i >> 5, mm = i & 31;
      int gk = k0 + kk, gm = m0 + mm;
      sX[kk][mm] = (gk < K && gm < M) ? g(b, gk, gm) : 0.f;
    }
#if HAVE_TDM
    if (wave == 0) __builtin_amdgcn_s_wait_tensorcnt(0);
#endif
    __syncthreads();
    wmma_chunk(sW, sX, nsub, msub, lane, acc);
    __syncthreads();
  }
  wmma_epilogue<RELU>(acc, Out + (size_t)b * out_bs,
                      bias ? bias + (size_t)b * bias_bs : nullptr,
                      n0, m0, nsub, msub, lane, N, M);
}

// ------------------------- elementwise / VALU kernels -----------------------

__global__ void pool_kernel(const float* __restrict__ in, float* __restrict__ out,
                            int sel) {
  int i = blockIdx.x * 256 + threadIdx.x;
  if (i >= BATCH * INC * HW) return;
  int m = i % HW; int r = i / HW; int c = r % INC; int b = r / INC;
  int y = m / WW, x = m % WW;
  const float* p = in + (((size_t)b * 2 + sel) * INC + c) * (size_t)(160 * 160)
                      + (size_t)(2 * y) * 160 + 2 * x;
  out[i] = 0.25f * (p[0] + p[1] + p[160] + p[161]);
}

__global__ void corr_kernel(const float* __restrict__ A, const float* __restrict__ Bc,
                            float* __restrict__ out) {
  int i = blockIdx.x * 256 + threadIdx.x;
  if (i >= BATCH * 49 * HW) return;
  int m = i % HW; int r = i / HW; int d = r % 49; int b = r / 49;
  int y = m / WW, x = m % WW;
  int yy = y + (d / 7) * 2 - 6;
  int xx = x + (d % 7) * 2 - 6;
  float s = 0.f;
  if ((unsigned)yy < (unsigned)HH && (unsigned)xx < (unsigned)WW) {
    const float* ap = A + (size_t)b * INC * HW + m;
    const float* bp = Bc + (size_t)b * INC * HW + yy * WW + xx;
#pragma unroll 4
    for (int c = 0; c < INC; ++c) s += ap[(size_t)c * HW] * bp[(size_t)c * HW];
  }
  out[i] = s * (1.f / 256.f);
}

__global__ void mean_kernel(const float* __restrict__ h, float* __restrict__ v) {
  int i = blockIdx.x * 256 + threadIdx.x;
  if (i >= BATCH * DIMS) return;
  const float* p = h + (size_t)i * 100;
  float s = 0.f;
#pragma unroll 10
  for (int j = 0; j < 100; ++j) s += p[j];
  v[i] = s * 0.01f;
}

__global__ void fc_kernel(const float* __restrict__ V, const float* __restrict__ W,
                          const float* __restrict__ bias, float* __restrict__ out,
                          int J) {
  int j = blockIdx.x * 256 + threadIdx.x;
  if (j >= J) return;
  const float* wr = W + (size_t)j * DIMS;
  float a0 = 0.f, a1 = 0.f, a2 = 0.f, a3 = 0.f;
#pragma unroll 4
  for (int k = 0; k < DIMS; ++k) {
    float wv = wr[k];
    a0 += V[0 * DIMS + k] * wv;
    a1 += V[1 * DIMS + k] * wv;
    a2 += V[2 * DIMS + k] * wv;
    a3 += V[3 * DIMS + k] * wv;
  }
  float bb = bias[j];
  out[(size_t)0 * J + j] = a0 + bb;
  out[(size_t)1 * J + j] = a1 + bb;
  out[(size_t)2 * J + j] = a2 + bb;
  out[(size_t)3 * J + j] = a3 + bb;
}

__global__ void sample_kernel(const float* __restrict__ aenc,
                              const float* __restrict__ off,
                              float* __restrict__ samp) {
  long long i = (long long)blockIdx.x * 256 + threadIdx.x;
  if (i >= (long long)BATCH * DIMS * 9 * HW) return;
  int m = (int)(i % HW); long long r = i / HW;
  int k = (int)(r % (DIMS * 9)); int b = (int)(r / (DIMS * 9));
  int c = k / 9, t = k - c * 9;
  const float* ob = off + (size_t)b * 18 * HW;
  float offy = ob[(size_t)(2 * t) * HW + m];
  float offx = ob[(size_t)(2 * t + 1) * HW + m];
  int y = m / WW, x = m % WW;
  float py = (float)(y + t / 3 - 1) + offy;
  float px = (float)(x + t % 3 - 1) + offx;
  float y0f = floorf(py), x0f = floorf(px);
  float wy = py - y0f, wx = px - x0f;
  int y0 = (int)y0f, x0 = (int)x0f;
  const float* img = aenc + ((size_t)b * DIMS + c) * HW;
  auto G = [&](int yi, int xi) -> float {
    if ((unsigned)yi >= (unsigned)HH || (unsigned)xi >= (unsigned)WW) return 0.f;
    return img[yi * WW + xi];
  };
  float v = G(y0, x0) * (1.f - wy) * (1.f - wx)
          + G(y0, x0 + 1) * (1.f - wy) * wx
          + G(y0 + 1, x0) * wy * (1.f - wx)
          + G(y0 + 1, x0 + 1) * wy * wx;
  samp[((size_t)b * DIMS * 9 + k) * HW + m] = v;
}

inline unsigned blocks1d(long long n) { return (unsigned)((n + 255) / 256); }

}  // namespace

// ---------------------------------------------------------------------------

extern "C" void kernel_launch(void* const* d_in, const int* in_sizes, int n_in,
                              void* d_out, int out_size, void* d_ws, size_t ws_size,
                              hipStream_t stream) {
  (void)in_sizes; (void)n_in; (void)out_size; (void)ws_size;
  // Flattened input order (jax pytree: sorted dict keys):
  // 0:R0 1:T0 2:inputs 3:enc0_b 4:enc0_w 5:enc1_b 6:enc1_w 7:fusion_w
  // 8:off0_w 9:off1_w 10..16:pw0{bx_b,bx_w,c1,c2,c3,wx_b,wx_w} 17..23:pw1{...}
  const float* R0      = (const float*)d_in[0];
  const float* T0      = (const float*)d_in[1];
  const float* inputs  = (const float*)d_in[2];
  const float* enc_b[2]  = {(const float*)d_in[3], (const float*)d_in[5]};
  const float* enc_w[2]  = {(const float*)d_in[4], (const float*)d_in[6]};
  const float* fusion_w  = (const float*)d_in[7];
  const float* off_w[2]  = {(const float*)d_in[8], (const float*)d_in[9]};
  const float* pw_bx_b[2] = {(const float*)d_in[10], (const float*)d_in[17]};
  const float* pw_bx_w[2] = {(const float*)d_in[11], (const float*)d_in[18]};
  const float* pw_c1[2]   = {(const float*)d_in[12], (const float*)d_in[19]};
  const float* pw_c2[2]   = {(const float*)d_in[13], (const float*)d_in[20]};
  const float* pw_c3[2]   = {(const float*)d_in[14], (const float*)d_in[21]};
  const float* pw_wx_b[2] = {(const float*)d_in[15], (const float*)d_in[22]};
  const float* pw_wx_w[2] = {(const float*)d_in[16], (const float*)d_in[23]};
  float* out = (float*)d_out;

  // ---- workspace carve (floats) ----
  float* ws = (float*)d_ws;
  auto alloc = [&](size_t n) { float* p = ws; ws += n; return p; };
  float* R_pre = alloc((size_t)BATCH * INC * HW);
  float* R_cur = alloc((size_t)BATCH * INC * HW);
  float* T_cur = alloc((size_t)BATCH * INC * HW);
  float* corr  = alloc((size_t)BATCH * 49 * HW);        // reused across streams
  float* offs  = alloc((size_t)BATCH * 18 * HW);
  float* h1    = alloc((size_t)BATCH * DIMS * 1600);
  float* h2    = alloc((size_t)BATCH * DIMS * 400);
  float* h3    = alloc((size_t)BATCH * DIMS * 100);
  float* vmean = alloc((size_t)BATCH * DIMS);
  float* wx    = alloc((size_t)BATCH * DIMS * DIMS * 9); // 147456 per batch
  float* bx    = alloc((size_t)BATCH * DIMS);
  float* aenc  = alloc((size_t)BATCH * DIMS * HW);
  float* samp  = alloc((size_t)BATCH * DIMS * 9 * HW);   // 118 MB, reused
  float* outs[2];
  outs[0] = alloc((size_t)BATCH * DIMS * HW);
  outs[1] = alloc((size_t)BATCH * DIMS * HW);

  const size_t inp_bs = (size_t)2 * INC * HW;  // batch stride inside `inputs`
  const dim3 blk(256);

  // ---- 1. pooling (only pass over the 315MB R0/T0 slices) ----
  pool_kernel<<<blocks1d((long long)BATCH * INC * HW), blk, 0, stream>>>(R0, R_pre, 0);
  pool_kernel<<<blocks1d((long long)BATCH * INC * HW), blk, 0, stream>>>(R0, R_cur, 1);
  pool_kernel<<<blocks1d((long long)BATCH * INC * HW), blk, 0, stream>>>(T0, T_cur, 1);

  for (int i = 0; i < 2; ++i) {
    const float* corrA = (i == 0) ? R_pre : R_cur;
    const float* a_base = inputs + (size_t)(i == 0 ? 0 : 1) * INC * HW;  // x / y
    const float* b_base = inputs + (size_t)(i == 0 ? 1 : 0) * INC * HW;  // y / x

    // ---- 2. correlation (operands L2-resident) ----
    corr_kernel<<<blocks1d((long long)BATCH * 49 * HW), blk, 0, stream>>>(
        corrA, T_cur, corr);

    // ---- 3. offset conv: 3x3 over concat(corr49, a256) -> 18ch ----
    {
      GConv3Pad g{corr, (size_t)49 * HW, 49, a_base, inp_bs, HH, WW};
      dim3 grid((HW + 31) / 32, (18 + 63) / 64, BATCH);
      gemm_wmma_g<GConv3Pad, false><<<grid, blk, 0, stream>>>(
          off_w[i], 0, nullptr, 0, offs, (size_t)18 * HW, 18, 305 * 9, HW, g);
    }
    // ---- 4a. c1: 5x5 s2 over concat(corr49, a256, b256), relu ----
    {
      GConvS2 g{corr, (size_t)49 * HW, 49, a_base, inp_bs, INC,
                b_base, inp_bs, 5, 2, HH, WW, 40};
      dim3 grid((1600 + 31) / 32, (DIMS + 63) / 64, BATCH);
      gemm_wmma_g<GConvS2, true><<<grid, blk, 0, stream>>>(
          pw_c1[i], 0, nullptr, 0, h1, (size_t)DIMS * 1600, DIMS, 561 * 25, 1600, g);
    }
    // ---- 4b. c2: 5x5 s2 128->128, relu ----
    {
      GConvS2 g{h1, (size_t)DIMS * 1600, DIMS, h1, 0, 0, h1, 0, 5, 2, 40, 40, 20};
      dim3 grid((400 + 31) / 32, (DIMS + 63) / 64, BATCH);
      gemm_wmma_g<GConvS2, true><<<grid, blk, 0, stream>>>(
          pw_c2[i], 0, nullptr, 0, h2, (size_t)DIMS * 400, DIMS, DIMS * 25, 400, g);
    }
    // ---- 4c. c3: 3x3 s2 128->128 ----
    {
      GConvS2 g{h2, (size_t)DIMS * 400, DIMS, h2, 0, 0, h2, 0, 3, 1, 20, 20, 10};
      dim3 grid((100 + 31) / 32, (DIMS + 63) / 64, BATCH);
      gemm_wmma_g<GConvS2, false><<<grid, blk, 0, stream>>>(
          pw_c3[i], 0, nullptr, 0, h3, (size_t)DIMS * 100, DIMS, DIMS * 9, 100, g);
    }
    // ---- 4d. GAP + FCs (wx FC = 75MB weight read: pure bandwidth) ----
    mean_kernel<<<blocks1d(BATCH * DIMS), blk, 0, stream>>>(h3, vmean);
    fc_kernel<<<blocks1d(DIMS * DIMS * 9), blk, 0, stream>>>(
        vmean, pw_wx_w[i], pw_wx_b[i], wx, DIMS * DIMS * 9);
    fc_kernel<<<blocks1d(DIMS), blk, 0, stream>>>(
        vmean, pw_bx_w[i], pw_bx_b[i], bx, DIMS);

    // ---- 5. encoder 1x1 conv: plain GEMM, full TDM staging ----
    {
      dim3 grid((HW + 31) / 32, (DIMS + 63) / 64, BATCH);
      gemm_wmma_plain<false><<<grid, blk, 0, stream>>>(
          enc_w[i], 0, a_base, inp_bs, enc_b[i], 0,
          aenc, (size_t)DIMS * HW, DIMS, INC, HW);
    }
    // ---- 6. deformable bilinear sampling -> samp[b][c*9+t][m] ----
    sample_kernel<<<blocks1d((long long)BATCH * DIMS * 9 * HW), blk, 0, stream>>>(
        aenc, offs, samp);
    // ---- 7. per-batch dynamic GEMM: wx[b](128x1152) x samp + bx, TDM ----
    {
      dim3 grid((HW + 31) / 32, (DIMS + 63) / 64, BATCH);
      gemm_wmma_plain<false><<<grid, blk, 0, stream>>>(
          wx, (size_t)DIMS * DIMS * 9, samp, (size_t)DIMS * 9 * HW, bx, DIMS,
          outs[i], (size_t)DIMS * HW, DIMS, DIMS * 9, HW);
    }
  }

  // ---- 8. fusion: 3x3 conv over concat(out0,out1) -> d_out (NCHW) ----
  {
    GConv3Pad g{outs[0], (size_t)DIMS * HW, DIMS, outs[1], (size_t)DIMS * HW, HH, WW};
    dim3 grid((HW + 31) / 32, (INC + 63) / 64, BATCH);
    gemm_wmma_g<GConv3Pad, false><<<grid, blk, 0, stream>>>(
        fusion_w, 0, nullptr, 0, out, (size_t)INC * HW, INC, 2 * DIMS * 9, HW, g);
  }
}